// SoftPolygon_2705829396670
// MI455X (gfx1250) — compile-verified
//
#include <hip/hip_runtime.h>

typedef __attribute__((ext_vector_type(2))) float v2f;
typedef __attribute__((ext_vector_type(8))) float v8f;

#define INV_SMOOTH 0.1f
#define POLY_EPS 1e-5f

// One block = one (batch, image-row). 256 threads = 8 waves (wave32).
// Wave w handles pixels x in [16*w, 16*w+16) of that row.
// Per wave: 8 chunks of 16 edges; each chunk = one V_WMMA_F32_16X16X4_F32
// computing tnum[pixel, edge] = px*dx + py*dy + c, then a VALU epilogue for
// clamp/projection/distance/crossing, then a 16-lane butterfly reduction
// over the edge columns.
__global__ __launch_bounds__(256) void softpoly_wmma_kernel(
    const float* __restrict__ vertices,   // [B,128,2]
    float* __restrict__ out,              // [B,H,W]
    int H, int W)
{
    const int blk  = blockIdx.x;
    const int b    = blk >> 7;          // / 128 rows
    const int row  = blk & 127;
    const int tid  = (int)threadIdx.x;
    const int lane = tid & 31;
    const int wave = tid >> 5;
    const int lane16 = lane & 15;
    const int hi     = lane >> 4;       // 0 for lanes 0-15, 1 for lanes 16-31

    // Packed edge parameters: one ds_load_b128 + one ds_load_b64 per chunk.
    __shared__ float4 s_e0[128];        // (ax, ay, dx, dy)
    __shared__ float2 s_e1[128];        // (c, invden)

    // ---- per-block edge preprocessing into LDS (shared by all 8 waves) ----
    if (tid < 128) {
        const int i = tid;
        const int j = (tid + 1) & 127;              // roll(-1)
        const float ax = vertices[((b * 128 + i) << 1) + 0];
        const float ay = vertices[((b * 128 + i) << 1) + 1];
        const float bx = vertices[((b * 128 + j) << 1) + 0];
        const float by = vertices[((b * 128 + j) << 1) + 1];
        const float dx = bx - ax;
        const float dy = by - ay;
        s_e0[i] = make_float4(ax, ay, dx, dy);
        s_e1[i] = make_float2(-(ax * dx + ay * dy),
                              1.0f / (dx * dx + dy * dy + POLY_EPS));
    }
    __syncthreads();

    const int   xbase = wave * 16;
    const float py    = (float)row;

    // A-matrix (16x4 f32, loop invariant):
    //   lanes 0-15 : VGPR0 = K0 = px(m=lane), VGPR1 = K1 = py
    //   lanes 16-31: VGPR0 = K2 = 1.0,        VGPR1 = K3 = 0.0
    v2f A;
    A.x = hi ? 1.0f : (float)(xbase + lane);
    A.y = hi ? 0.0f : py;

    float dmin[8];
    int   cnt[8];
#pragma unroll
    for (int v = 0; v < 8; ++v) { dmin[v] = 3.0e38f; cnt[v] = 0; }

#pragma unroll
    for (int e = 0; e < 8; ++e) {
        const int n = e * 16 + lane16;     // edge column owned by this lane
        const float4 e0 = s_e0[n];         // ds_load_b128
        const float2 e1 = s_e1[n];         // ds_load_b64
        const float ax = e0.x, ay = e0.y;
        const float dx = e0.z, dy = e0.w;
        const float c  = e1.x;
        const float invden = e1.y;

        // B-matrix (4x16 f32):
        //   lanes 0-15 : VGPR0 = K0 row = dx(n), VGPR1 = K1 row = dy(n)
        //   lanes 16-31: VGPR0 = K2 row = c(n),  VGPR1 = K3 row = 0
        v2f Bm;
        Bm.x = hi ? c    : dx;
        Bm.y = hi ? 0.0f : dy;

        v8f acc = {};
#if __has_builtin(__builtin_amdgcn_wmma_f32_16x16x4_f32)
        acc = __builtin_amdgcn_wmma_f32_16x16x4_f32(
            /*neg_a=*/false, A, /*neg_b=*/false, Bm,
            /*c_mod=*/(short)0, acc, /*reuse_a=*/false, /*reuse_b=*/false);
#else
        // VALU fallback with identical element layout
#pragma unroll
        for (int v = 0; v < 8; ++v) {
            const float pxf = (float)(xbase + v + 8 * hi);
            acc[v] = pxf * dx + py * dy + c;
        }
#endif

        // Scanline crossing precompute: depends only on (edge, row),
        // identical for all 16 pixels of this strip. cond => dy != 0, so a
        // fast reciprocal is safe (feeds a comparison only).
        const float byv   = ay + dy;
        const bool  cond  = (ay > py) != (byv > py);
        const float dysafe = cond ? dy : 1.0f;
        const float xint   = fmaf(dx * (py - ay), __frcp_rn(dysafe), ax);

#pragma unroll
        for (int v = 0; v < 8; ++v) {
            // element (M = v + 8*hi, N = lane16) of the 16x16 D tile
            const float px = (float)(xbase + v + 8 * hi);
            float t = acc[v] * invden;
            t = fminf(fmaxf(t, 0.0f), 1.0f);
            const float ex = fmaf(t, dx, ax);
            const float ey = fmaf(t, dy, ay);
            const float qx = px - ex;
            const float qy = py - ey;
            const float d2 = fmaf(qx, qx, qy * qy);
            dmin[v] = fminf(dmin[v], d2);
            cnt[v] += (cond && (px < xint)) ? 1 : 0;
        }
    }

    // ---- reduce over the 16 edge-columns (lanes of each half-wave) ----
#pragma unroll
    for (int v = 0; v < 8; ++v) {
#pragma unroll
        for (int m = 1; m < 16; m <<= 1) {
            dmin[v] = fminf(dmin[v], __shfl_xor(dmin[v], m, 32));
            cnt[v] += __shfl_xor(cnt[v], m, 32);
        }
    }

    // lane 0 holds pixels M=0..7 (v index), lane 16 holds M=8..15
    if (lane16 == 0) {
        float* orow = out + ((size_t)b * H + row) * W + xbase + hi * 8;
#pragma unroll
        for (int v = 0; v < 8; ++v) {
            const float sign = (cnt[v] & 1) ? 1.0f : -1.0f;
            const float z = sign * dmin[v] * (1.0f / INV_SMOOTH);
            orow[v] = __frcp_rn(1.0f + __expf(-z));
        }
    }
}

extern "C" void kernel_launch(void* const* d_in, const int* in_sizes, int n_in,
                              void* d_out, int out_size, void* d_ws, size_t ws_size,
                              hipStream_t stream)
{
    (void)in_sizes; (void)n_in; (void)d_ws; (void)ws_size; (void)out_size;
    const float* vertices = (const float*)d_in[0];   // [32,128,2] f32
    float* out = (float*)d_out;                      // [32,128,128] f32

    const int B = 32, H = 128, W = 128;
    dim3 grid(B * H);        // one block per (batch, row)
    dim3 block(256);         // 8 waves, one 16-pixel strip each
    hipLaunchKernelGGL(softpoly_wmma_kernel, grid, block, 0, stream,
                       vertices, out, H, W);
}